// GCN_2104533975458
// MI455X (gfx1250) — compile-verified
//
#include <hip/hip_runtime.h>
#include <hip/hip_bf16.h>

// ---------------------------------------------------------------------------
// GCN layer with spectral-normalized weight, sigmoid edge weights,
// self-loops + symmetric normalization.
//
// Pipeline (all on `stream`):
//   1) spectral_kernel      : W_sn = W / sigma                (1 block)
//   2) init_deg_kernel      : deg[i] = 1 (self loop)
//   3) edge_kernel          : w[e]=sigmoid(ewt[e]); deg[col]+=w (atomics)
//   4) dinv_kernel          : dinv = rsqrt(deg)
//   5) gemm_wmma_kernel     : xw = x @ W_sn   (V_WMMA_F32_16X16X4_F32)
//   6) out_init_kernel      : out = bias + (dinv^2)*xw   (self-loop term)
//   7) scatter_kernel       : out[col] += dinv[row]*w*dinv[col] * xw[row]
// ---------------------------------------------------------------------------

typedef __attribute__((ext_vector_type(2))) float v2f;
typedef __attribute__((ext_vector_type(8))) float v8f;

#define FDIM 128

// ---------------- 1) spectral norm: W_sn = W / sigma -----------------------
// sigma = ||W v|| (with torch-style eps), v = l2norm(W^T u).
__global__ __launch_bounds__(128) void spectral_kernel(const float* __restrict__ W,
                                                       const float* __restrict__ u,
                                                       float* __restrict__ Wsn) {
    __shared__ float sv[FDIM];
    __shared__ float red[FDIM];
    const int t = threadIdx.x;  // 0..127

    // v_t = sum_k W[k][t] * u[k]
    float acc = 0.f;
    for (int k = 0; k < FDIM; ++k) acc += W[k * FDIM + t] * u[k];

    red[t] = acc * acc;
    __syncthreads();
    for (int s = 64; s > 0; s >>= 1) {
        if (t < s) red[t] += red[t + s];
        __syncthreads();
    }
    const float vnorm = sqrtf(red[0]) + 1e-12f;
    __syncthreads();
    sv[t] = acc / vnorm;
    __syncthreads();

    // Wv_t = sum_k W[t][k] * v[k]
    float wv = 0.f;
    for (int k = 0; k < FDIM; ++k) wv += W[t * FDIM + k] * sv[k];

    red[t] = wv * wv;
    __syncthreads();
    for (int s = 64; s > 0; s >>= 1) {
        if (t < s) red[t] += red[t + s];
        __syncthreads();
    }
    // sigma = u1 . Wv with u1 = Wv/(||Wv||+eps)  ->  ||Wv||^2 / (||Wv||+eps)
    const float wvsq = red[0];
    const float sigma = wvsq / (sqrtf(wvsq) + 1e-12f);
    const float inv_sigma = 1.0f / sigma;

    for (int i = t; i < FDIM * FDIM; i += 128) Wsn[i] = W[i] * inv_sigma;
}

// ---------------- 2) deg init (self loop contributes 1) --------------------
__global__ void init_deg_kernel(float* __restrict__ deg, int N) {
    int i = blockIdx.x * blockDim.x + threadIdx.x;
    if (i < N) deg[i] = 1.0f;
}

// ---------------- 3) edge sigmoid + degree accumulation --------------------
__global__ void edge_kernel(const float* __restrict__ ewt,
                            const int* __restrict__ col,
                            float* __restrict__ w,
                            float* __restrict__ deg, int E) {
    int e = blockIdx.x * blockDim.x + threadIdx.x;
    if (e >= E) return;
    float s = 1.0f / (1.0f + __expf(-ewt[e]));
    w[e] = s;
    unsafeAtomicAdd(&deg[col[e]], s);
}

// ---------------- 4) dinv = deg^-1/2 (deg >= 1 always) ---------------------
__global__ void dinv_kernel(float* __restrict__ deg, int N) {
    int i = blockIdx.x * blockDim.x + threadIdx.x;
    if (i < N) {
        float d = deg[i];
        deg[i] = (d > 0.f) ? rsqrtf(d) : 0.f;  // in-place: deg becomes dinv
    }
}

// ---------------- 5) xw = x @ W_sn via V_WMMA_F32_16X16X4_F32 --------------
// One block = 8 waves = one 16-row stripe; wave w owns output cols [16w,16w+16).
// W_sn staged in LDS (64 KB) once per block and reused by all 8 waves.
__global__ __launch_bounds__(256) void gemm_wmma_kernel(const float* __restrict__ x,
                                                        const float* __restrict__ Wsn,
                                                        float* __restrict__ xw, int N) {
    __shared__ float sW[FDIM * FDIM];  // 64 KB
    const int tid = threadIdx.x;

    // cooperative 64 KB load: 256 threads * 16 x float4
    for (int i = tid * 4; i < FDIM * FDIM; i += 256 * 4) {
        *(float4*)(sW + i) = *(const float4*)(Wsn + i);
    }
    __syncthreads();

    const int wave = tid >> 5;
    const int lane = tid & 31;
    const int m0 = blockIdx.x * 16;
    const int col0 = wave * 16;

    const int l = lane & 15;           // N index within tile / M index for A rows
    const int koff = (lane >> 4) << 1; // lanes 0-15 -> K+0,K+1 ; lanes 16-31 -> K+2,K+3

    int arow = m0 + l;
    if (arow >= N) arow = N - 1;       // clamp (N is a multiple of 16 anyway)
    const float* __restrict__ xr = x + (size_t)arow * FDIM + koff;
    const float* __restrict__ bp = sW + koff * FDIM + col0 + l;

    v8f c = {};
#pragma unroll
    for (int kb = 0; kb < FDIM; kb += 4) {
        // A fragment (16x4 f32): lane l<16: {X[m0+l][kb], X[m0+l][kb+1]}, l>=16: K+2/K+3
        float2 av = *(const float2*)(xr + kb);
        v2f a;  a[0] = av.x; a[1] = av.y;
        // B fragment (4x16 f32): VGPR0 = W[kb+koff][col0+l], VGPR1 = W[kb+koff+1][col0+l]
        v2f b;  b[0] = bp[kb * FDIM]; b[1] = bp[(kb + 1) * FDIM];
        c = __builtin_amdgcn_wmma_f32_16x16x4_f32(
                /*neg_a=*/false, a, /*neg_b=*/false, b,
                /*c_mod=*/(short)0, c, /*reuse_a=*/false, /*reuse_b=*/false);
    }

    // C/D layout: VGPR r -> M = m0 + r (+8 for lanes 16-31), N = col0 + l
    const int mbase = m0 + ((lane >> 4) << 3);
#pragma unroll
    for (int r = 0; r < 8; ++r) {
        int m = mbase + r;
        if (m < N) xw[(size_t)m * FDIM + col0 + l] = c[r];
    }
}

// ---------------- 6) out = bias + dinv^2 * xw (self-loop term) -------------
__global__ void out_init_kernel(const float* __restrict__ xw,
                                const float* __restrict__ dinv,
                                const float* __restrict__ bias,
                                float* __restrict__ out, int total) {
    int t = blockIdx.x * blockDim.x + threadIdx.x;
    if (t >= total) return;
    int i = t >> 7;        // node
    int f = t & (FDIM - 1);
    float di = dinv[i];
    out[t] = bias[f] + di * di * xw[t];
}

// ---------------- 7) per-edge gather + atomic scatter ----------------------
// One wave per edge; each lane moves 4 consecutive floats (32*4 = 128).
__global__ __launch_bounds__(256) void scatter_kernel(const float* __restrict__ xw,
                                                      const float* __restrict__ dinv,
                                                      const float* __restrict__ w,
                                                      const int* __restrict__ row,
                                                      const int* __restrict__ col,
                                                      float* __restrict__ out, int E) {
    int gid = blockIdx.x * blockDim.x + threadIdx.x;
    int e = gid >> 5;
    int lane = gid & 31;
    if (e >= E) return;

    int r = row[e];
    int c = col[e];
    float coeff = dinv[r] * w[e] * dinv[c];

    const float4 xv = *(const float4*)(xw + (size_t)r * FDIM + lane * 4);
    float* op = out + (size_t)c * FDIM + lane * 4;
    unsafeAtomicAdd(op + 0, coeff * xv.x);
    unsafeAtomicAdd(op + 1, coeff * xv.y);
    unsafeAtomicAdd(op + 2, coeff * xv.z);
    unsafeAtomicAdd(op + 3, coeff * xv.w);
}

// ---------------------------------------------------------------------------
extern "C" void kernel_launch(void* const* d_in, const int* in_sizes, int n_in,
                              void* d_out, int out_size, void* d_ws, size_t ws_size,
                              hipStream_t stream) {
    const float* x    = (const float*)d_in[0];   // [N,128]
    const int*   ei   = (const int*)d_in[1];     // [2,E]
    const float* ewt  = (const float*)d_in[2];   // [E]
    const float* Wgt  = (const float*)d_in[3];   // [128,128]
    const float* bias = (const float*)d_in[4];   // [128]
    const float* u    = (const float*)d_in[5];   // [128]
    float* out = (float*)d_out;

    const int N = in_sizes[0] / FDIM;
    const int E = in_sizes[2];
    const int* row = ei;
    const int* col = ei + E;

    // workspace layout (floats): Wsn[16384] | dinv[N] | w[E] | xw[N*128]
    float* ws   = (float*)d_ws;
    float* Wsn  = ws;
    float* dinv = Wsn + FDIM * FDIM;
    float* wbuf = dinv + N;
    float* xw   = wbuf + E;

    spectral_kernel<<<1, 128, 0, stream>>>(Wgt, u, Wsn);

    init_deg_kernel<<<(N + 255) / 256, 256, 0, stream>>>(dinv, N);
    edge_kernel<<<(E + 255) / 256, 256, 0, stream>>>(ewt, col, wbuf, dinv, E);
    dinv_kernel<<<(N + 255) / 256, 256, 0, stream>>>(dinv, N);

    gemm_wmma_kernel<<<(N + 15) / 16, 256, 0, stream>>>(x, Wsn, xw, N);

    const int total = N * FDIM;
    out_init_kernel<<<(total + 255) / 256, 256, 0, stream>>>(xw, dinv, bias, out, total);

    // 32 lanes per edge
    long long sthreads = (long long)E * 32;
    scatter_kernel<<<(int)((sthreads + 255) / 256), 256, 0, stream>>>(
        xw, dinv, wbuf, row, col, out, E);
}